// LightGCN_32581621907927
// MI455X (gfx1250) — compile-verified
//
#include <hip/hip_runtime.h>
#include <stdint.h>

#define UCNT   100000
#define ICNT   50000
#define NNODES 150000
#define DIM    64
#define NE     3200000
#define NB     8192
#define NLAYER 4           // L+1 embedding layers

typedef __attribute__((ext_vector_type(2))) float v2f;
typedef __attribute__((ext_vector_type(8))) float v8f;

// ---------------------------------------------------------------- utilities
__device__ __forceinline__ uint32_t mix32(uint32_t x) {
    x ^= x >> 16; x *= 0x7FEB352Du;
    x ^= x >> 15; x *= 0x846CA68Bu;
    x ^= x >> 16;
    return x;
}

// Deterministic standard-normal stand-in for jax.random.normal(fold_in(key,i))
__device__ __forceinline__ float gaussian(uint32_t layer, uint32_t row, uint32_t col) {
    uint32_t s  = mix32(layer * 0x9E3779B9u + row * 0x85EBCA6Bu + col * 0xC2B2AE35u + 0x5bd1e995u);
    uint32_t h1 = mix32(s ^ 0x68bc21ebu);
    uint32_t h2 = mix32(s ^ 0x02e5be93u);
    float u1 = ((h1 >> 8) + 1u) * (1.0f / 16777217.0f);   // (0,1]
    float u2 = (h2 >> 8) * (1.0f / 16777216.0f);          // [0,1)
    float r  = sqrtf(-2.0f * __logf(u1));
    return r * __cosf(6.28318530718f * u2);
}

// ---------------------------------------------------------------- init layer0
__global__ __launch_bounds__(256) void k_init_embs0(const float* __restrict__ ut,
                                                    const float* __restrict__ it,
                                                    float* __restrict__ e0) {
    long long t = (long long)blockIdx.x * blockDim.x + threadIdx.x;   // one float4 each
    long long total = (long long)NNODES * (DIM / 4);
    if (t >= total) return;
    long long node = t >> 4, q = t & 15;
    const float4* src = (node < UCNT)
        ? (const float4*)ut + node * 16 + q
        : (const float4*)it + (node - UCNT) * 16 + q;
    ((float4*)e0)[node * 16 + q] = *src;
}

// ---------------------------------------------------------------- degree / norm
__global__ __launch_bounds__(256) void k_degree(const int* __restrict__ src,
                                                float* __restrict__ deg) {
    int e = blockIdx.x * blockDim.x + threadIdx.x;
    if (e < NE) atomicAdd(&deg[src[e]], 1.0f);
}

__global__ __launch_bounds__(256) void k_dinv(float* __restrict__ deg) {
    int i = blockIdx.x * blockDim.x + threadIdx.x;
    if (i < NNODES) deg[i] = rsqrtf(deg[i] + 1e-10f);
}

// ---------------------------------------------------------------- SpMM (wave per edge)
__global__ __launch_bounds__(256) void k_spmm(const int*   __restrict__ src,
                                              const int*   __restrict__ dst,
                                              const float* __restrict__ dinv,
                                              const float* __restrict__ x,
                                              float*       __restrict__ out) {
    long long t = (long long)blockIdx.x * blockDim.x + threadIdx.x;
    long long e = t >> 5;                 // 32 lanes share one edge
    if (e >= NE) return;
    int lane = threadIdx.x & 31;
    int s = src[e], d = dst[e];
    float v = dinv[s] * dinv[d];
    float2 xx = *(const float2*)&x[(size_t)d * DIM + 2 * lane];
    float* o = &out[(size_t)s * DIM + 2 * lane];
    atomicAdd(o,     v * xx.x);
    atomicAdd(o + 1, v * xx.y);
}

// ---------------------------------------------------------------- fused VAE loss (WMMA f32)
// One wave owns a 16-row tile of x.  mu = xW_mu+b, lv = xW_lv+b via
// V_WMMA_F32_16X16X4_F32 (A frags register-resident, reused for both GEMMs),
// z round-trips through per-wave LDS to re-enter A layout for the decoder GEMM.
__global__ __launch_bounds__(128) void k_vae(const float* __restrict__ x,
                                             const float* __restrict__ Wmu,
                                             const float* __restrict__ bmu,
                                             const float* __restrict__ Wlv,
                                             const float* __restrict__ blv,
                                             const float* __restrict__ Wdec,
                                             const float* __restrict__ bdec,
                                             float* __restrict__ acc, int layer) {
    __shared__ float zt[4][16 * DIM];                 // 4 KB per wave
    const int lane = threadIdx.x & 31;
    const int wave = threadIdx.x >> 5;
    const int tile = blockIdx.x * 4 + wave;
    const int row0 = tile * 16;                       // 150000/16 = 9375 exact
    const bool valid = row0 < NNODES;                 // wave-uniform
    const int half = lane >> 4;                       // 0: K={0,1}, 1: K={2,3}
    const int m    = lane & 15;

    v2f afrag[16];
    float ssum = 0.f, rsum = 0.f;

    if (valid) {
        // ---- A fragments: x tile, 16 k-steps of 4 -------------------------
        #pragma unroll
        for (int kk = 0; kk < 16; ++kk) {
            int k = kk * 4 + 2 * half;
            float2 t = *(const float2*)&x[(size_t)(row0 + m) * DIM + k];
            afrag[kk].x = t.x; afrag[kk].y = t.y;
        }
        // ---- mu and logvar GEMMs (share A frags) --------------------------
        #pragma unroll
        for (int n = 0; n < 4; ++n) {                 // 4 column tiles of 16
            v8f cmu = {0.f,0.f,0.f,0.f,0.f,0.f,0.f,0.f};
            v8f clv = {0.f,0.f,0.f,0.f,0.f,0.f,0.f,0.f};
            #pragma unroll
            for (int kk = 0; kk < 16; ++kk) {
                int k = kk * 4 + 2 * half;
                v2f bm, bl;
                bm.x = Wmu[(size_t)(k    ) * DIM + n * 16 + m];
                bm.y = Wmu[(size_t)(k + 1) * DIM + n * 16 + m];
                bl.x = Wlv[(size_t)(k    ) * DIM + n * 16 + m];
                bl.y = Wlv[(size_t)(k + 1) * DIM + n * 16 + m];
                cmu = __builtin_amdgcn_wmma_f32_16x16x4_f32(false, afrag[kk], false, bm,
                                                            (short)0, cmu, false, false);
                clv = __builtin_amdgcn_wmma_f32_16x16x4_f32(false, afrag[kk], false, bl,
                                                            (short)0, clv, false, false);
            }
            // ---- bias, KL partial, z = mu + eps*exp(0.5*lv) ---------------
            int col = n * 16 + m;
            float bmub = bmu[col], blvb = blv[col];
            #pragma unroll
            for (int r = 0; r < 8; ++r) {
                int M = r + 8 * half;                 // C-layout row
                float mue = cmu[r] + bmub;
                float lve = clv[r] + blvb;
                ssum += lve - mue * mue - __expf(lve);
                float eps = gaussian((uint32_t)layer, (uint32_t)(row0 + M), (uint32_t)col);
                zt[wave][M * DIM + col] = mue + eps * __expf(0.5f * lve);
            }
        }
    }
    __syncthreads();
    if (valid) {
        // ---- decoder GEMM: recon = z @ W_dec ------------------------------
        #pragma unroll
        for (int kk = 0; kk < 16; ++kk) {
            int k = kk * 4 + 2 * half;
            afrag[kk].x = zt[wave][m * DIM + k];
            afrag[kk].y = zt[wave][m * DIM + k + 1];
        }
        #pragma unroll
        for (int n = 0; n < 4; ++n) {
            v8f c = {0.f,0.f,0.f,0.f,0.f,0.f,0.f,0.f};
            #pragma unroll
            for (int kk = 0; kk < 16; ++kk) {
                int k = kk * 4 + 2 * half;
                v2f bd;
                bd.x = Wdec[(size_t)(k    ) * DIM + n * 16 + m];
                bd.y = Wdec[(size_t)(k + 1) * DIM + n * 16 + m];
                c = __builtin_amdgcn_wmma_f32_16x16x4_f32(false, afrag[kk], false, bd,
                                                          (short)0, c, false, false);
            }
            int col = n * 16 + m;
            float bdb = bdec[col];
            #pragma unroll
            for (int r = 0; r < 8; ++r) {
                int M = r + 8 * half;
                float rec = c[r] + bdb;
                float xe  = x[(size_t)(row0 + M) * DIM + col];
                float df  = rec - xe;
                rsum += df * df;
            }
        }
    }
    // ---- wave reduction + single atomic per wave --------------------------
    #pragma unroll
    for (int o = 16; o > 0; o >>= 1) {
        rsum += __shfl_xor(rsum, o, 32);
        ssum += __shfl_xor(ssum, o, 32);
    }
    if (valid && lane == 0) {
        atomicAdd(&acc[layer],     rsum);   // sum (recon-x)^2
        atomicAdd(&acc[4 + layer], ssum);   // sum (lv - mu^2 - exp(lv))
    }
}

// ---------------------------------------------------------------- attention combine
__global__ __launch_bounds__(256) void k_attn(const float* __restrict__ embs,
                                              const float* __restrict__ aW,
                                              const float* __restrict__ ab,
                                              float* __restrict__ fin) {
    int lane = threadIdx.x & 31;
    int node = blockIdx.x * (blockDim.x >> 5) + (threadIdx.x >> 5);
    if (node >= NNODES) return;
    int d0 = lane * 2;
    float e[NLAYER][2];
    float p[NLAYER] = {0.f, 0.f, 0.f, 0.f};
    #pragma unroll
    for (int l = 0; l < NLAYER; ++l) {
        float2 t = *(const float2*)&embs[((size_t)l * NNODES + node) * DIM + d0];
        e[l][0] = t.x; e[l][1] = t.y;
        #pragma unroll
        for (int j = 0; j < NLAYER; ++j)
            p[j] += t.x * aW[(l * DIM + d0) * NLAYER + j] +
                    t.y * aW[(l * DIM + d0 + 1) * NLAYER + j];
    }
    #pragma unroll
    for (int o = 16; o > 0; o >>= 1)
        #pragma unroll
        for (int j = 0; j < NLAYER; ++j) p[j] += __shfl_xor(p[j], o, 32);
    float mx = -3.4e38f;
    #pragma unroll
    for (int j = 0; j < NLAYER; ++j) { p[j] += ab[j]; mx = fmaxf(mx, p[j]); }
    float sum = 0.f;
    #pragma unroll
    for (int j = 0; j < NLAYER; ++j) { p[j] = __expf(p[j] - mx); sum += p[j]; }
    float inv = 1.0f / sum;
    float o0 = 0.f, o1 = 0.f;
    #pragma unroll
    for (int l = 0; l < NLAYER; ++l) {
        float a = p[l] * inv;
        o0 += a * e[l][0]; o1 += a * e[l][1];
    }
    float2 outv; outv.x = o0; outv.y = o1;
    *(float2*)&fin[(size_t)node * DIM + d0] = outv;
}

// ---------------------------------------------------------------- scoring (wave per pair)
__global__ __launch_bounds__(256) void k_score(const float* __restrict__ fin,
                                               const int* __restrict__ users,
                                               const int* __restrict__ items,
                                               float* __restrict__ out) {
    int lane = threadIdx.x & 31;
    int b = blockIdx.x * (blockDim.x >> 5) + (threadIdx.x >> 5);
    if (b >= NB) return;
    int u = users[b];
    int v = UCNT + items[b];
    float2 a = *(const float2*)&fin[(size_t)u * DIM + 2 * lane];
    float2 c = *(const float2*)&fin[(size_t)v * DIM + 2 * lane];
    float s = a.x * c.x + a.y * c.y;
    #pragma unroll
    for (int o = 16; o > 0; o >>= 1) s += __shfl_xor(s, o, 32);
    if (lane == 0) out[b] = s;
}

// ---------------------------------------------------------------- final scalar loss
__global__ void k_final_loss(const float* __restrict__ acc, float* __restrict__ out) {
    if (threadIdx.x == 0 && blockIdx.x == 0) {
        const float inv = 1.0f / ((float)NNODES * (float)DIM);
        float tot = 0.f;
        for (int i = 0; i < 4; ++i) {
            float recon = acc[i] * inv;
            float kl    = -0.5f * (1.0f + acc[4 + i] * inv);
            tot += recon + kl;
        }
        out[NB] = tot * 0.25f;
    }
}

// ---------------------------------------------------------------- launcher
extern "C" void kernel_launch(void* const* d_in, const int* in_sizes, int n_in,
                              void* d_out, int out_size, void* d_ws, size_t ws_size,
                              hipStream_t stream) {
    const float* user_table = (const float*)d_in[0];
    const float* item_table = (const float*)d_in[1];
    const float* W_mu  = (const float*)d_in[2];
    const float* b_mu  = (const float*)d_in[3];
    const float* W_lv  = (const float*)d_in[4];
    const float* b_lv  = (const float*)d_in[5];
    const float* W_dec = (const float*)d_in[6];
    const float* b_dec = (const float*)d_in[7];
    const float* attn_W = (const float*)d_in[8];
    const float* attn_b = (const float*)d_in[9];
    const int* edge_src = (const int*)d_in[10];
    const int* edge_dst = (const int*)d_in[11];
    const int* users    = (const int*)d_in[12];
    const int* items    = (const int*)d_in[13];

    float* ws   = (float*)d_ws;
    float* embs = ws;                                     // 4 * N * D
    float* dinv = embs + (size_t)NLAYER * NNODES * DIM;   // N (rowsum then d^-1/2)
    float* fin  = dinv + NNODES;                          // N * D
    float* acc  = fin + (size_t)NNODES * DIM;             // 8 loss accumulators

    hipMemsetAsync(embs + (size_t)NNODES * DIM, 0,
                   (size_t)3 * NNODES * DIM * sizeof(float), stream);
    hipMemsetAsync(dinv, 0, (size_t)NNODES * sizeof(float), stream);
    hipMemsetAsync(acc, 0, 8 * sizeof(float), stream);

    {
        long long total = (long long)NNODES * (DIM / 4);
        k_init_embs0<<<(int)((total + 255) / 256), 256, 0, stream>>>(user_table, item_table, embs);
    }
    k_degree<<<(NE + 255) / 256, 256, 0, stream>>>(edge_src, dinv);
    k_dinv<<<(NNODES + 255) / 256, 256, 0, stream>>>(dinv);

    for (int l = 0; l < 3; ++l) {
        long long total = (long long)NE * 32;
        k_spmm<<<(int)((total + 255) / 256), 256, 0, stream>>>(
            edge_src, edge_dst, dinv,
            embs + (size_t)l * NNODES * DIM,
            embs + (size_t)(l + 1) * NNODES * DIM);
    }

    const int ntiles = NNODES / 16;                       // 9375, exact
    const int vae_blocks = (ntiles + 3) / 4;
    for (int i = 0; i < 4; ++i) {
        const float* x = embs + (size_t)((i < 3) ? (i + 1) : 3) * NNODES * DIM;
        k_vae<<<vae_blocks, 128, 0, stream>>>(
            x, W_mu + (size_t)i * DIM * DIM, b_mu + (size_t)i * DIM,
            W_lv + (size_t)i * DIM * DIM, b_lv + (size_t)i * DIM,
            W_dec + (size_t)i * DIM * DIM, b_dec + (size_t)i * DIM, acc, i);
    }

    k_attn<<<(NNODES + 7) / 8, 256, 0, stream>>>(embs, attn_W, attn_b, fin);
    k_score<<<(NB + 7) / 8, 256, 0, stream>>>(fin, users, items, (float*)d_out);
    k_final_loss<<<1, 64, 0, stream>>>(acc, (float*)d_out);
}